// BaseGinNetwork_86672440033288
// MI455X (gfx1250) — compile-verified
//
#include <hip/hip_runtime.h>
#include <hip/hip_bf16.h>

// ---------------- problem constants ----------------
#define NT   50000
#define NV   1000
#define NN   51000        // NT + NV
#define E1N  200000
#define E2N  100000
#define EN   300000       // E1N + E2N
#define HID  256
#define EMB  128

typedef float v2f __attribute__((ext_vector_type(2)));
typedef float v8f __attribute__((ext_vector_type(8)));

// ---------------- small utility kernels ----------------

__global__ void k_fill0(float* __restrict__ p, long long n) {
    long long i = (long long)blockIdx.x * blockDim.x + threadIdx.x;
    if (i < n) p[i] = 0.0f;
}

// max over vm_cpu_cores, clamped to >= 1.0
__global__ void k_max(const float* __restrict__ v, int n, float* __restrict__ out) {
    __shared__ float red[256];
    float m = 1.0f;
    for (int i = threadIdx.x; i < n; i += 256) m = fmaxf(m, v[i]);
    red[threadIdx.x] = m;
    __syncthreads();
    for (int s = 128; s > 0; s >>= 1) {
        if (threadIdx.x < s) red[threadIdx.x] = fmaxf(red[threadIdx.x], red[threadIdx.x + s]);
        __syncthreads();
    }
    if (threadIdx.x == 0) out[0] = red[0];
}

// zero-pad first-layer weights:  te_W1 [6,256] -> [8,256], ve_W1 [10,256] -> [12,256]
__global__ void k_packw(const float* __restrict__ teW1, const float* __restrict__ veW1,
                        float* __restrict__ w1p, float* __restrict__ wv1p) {
    int idx = blockIdx.x * blockDim.x + threadIdx.x;
    if (idx < 8 * 256) {
        int k = idx >> 8, n = idx & 255;
        w1p[idx] = (k < 6) ? teW1[k * 256 + n] : 0.0f;
    }
    if (idx < 12 * 256) {
        int k = idx / 256, n = idx % 256;
        wv1p[idx] = (k < 10) ? veW1[k * 256 + n] : 0.0f;
    }
}

// build padded feature rows: task_x [NT,8], vm_x [NV,12]
__global__ void k_feat(const float* t0, const float* t1, const float* t2,
                       const float* t3, const float* t4, const float* t5,
                       const float* v0, const float* v1, const float* v2,
                       const float* v3, const float* v4, const float* v5,
                       const float* v6, const float* v7, const float* v8,
                       const float* v9,
                       float* __restrict__ taskx, float* __restrict__ vmx,
                       const float* __restrict__ maxcp) {
    long long i = (long long)blockIdx.x * blockDim.x + threadIdx.x;
    float maxc = maxcp[0];
    if (i < NT) {
        float* r = taskx + i * 8;
        r[0] = t0[i]; r[1] = t1[i]; r[2] = t2[i]; r[3] = t3[i];
        r[4] = t4[i] / 1000.0f; r[5] = t5[i] / maxc; r[6] = 0.0f; r[7] = 0.0f;
    }
    if (i < NV) {
        float* r = vmx + i * 12;
        r[0] = v0[i]; r[1] = 1.0f / (v1[i] + 1e-8f); r[2] = v2[i];
        r[3] = v3[i] / 1000.0f; r[4] = v4[i] / 1000.0f; r[5] = v5[i];
        r[6] = v6[i]; r[7] = v7[i] / maxc; r[8] = v8[i] / maxc; r[9] = v9[i];
        r[10] = 0.0f; r[11] = 0.0f;
    }
}

// per-column batchnorm stats over h [Mrows, C]; folds BN into affine s,t
__global__ void k_colstats(const float* __restrict__ h, int Mrows, int C,
                           const float* __restrict__ g, const float* __restrict__ be,
                           float* __restrict__ s, float* __restrict__ t) {
    __shared__ float rs[256], rq[256];
    int c = blockIdx.x;
    float sum = 0.0f, sq = 0.0f;
    for (int i = threadIdx.x; i < Mrows; i += 256) {
        float v = h[(size_t)i * C + c];
        sum += v; sq += v * v;
    }
    rs[threadIdx.x] = sum; rq[threadIdx.x] = sq;
    __syncthreads();
    for (int w = 128; w > 0; w >>= 1) {
        if (threadIdx.x < w) { rs[threadIdx.x] += rs[threadIdx.x + w]; rq[threadIdx.x] += rq[threadIdx.x + w]; }
        __syncthreads();
    }
    if (threadIdx.x == 0) {
        float mean = rs[0] / (float)Mrows;
        float var  = rq[0] / (float)Mrows - mean * mean;
        float ss = g[c] * rsqrtf(var + 1e-5f);
        s[c] = ss;
        t[c] = be[c] - mean * ss;
    }
}

// segment-sum scatter: agg[dst] += x[src] over all E edges, C columns (float4/thread)
__global__ void k_scatter(const float* __restrict__ x, float* __restrict__ agg,
                          const int* __restrict__ cs, const int* __restrict__ cd,
                          const int* __restrict__ dsrc, const int* __restrict__ ddst,
                          int C) {
    long long idx = (long long)blockIdx.x * blockDim.x + threadIdx.x;
    int vecs = C >> 2;
    long long total = (long long)EN * vecs;
    if (idx >= total) return;
    int e = (int)(idx / vecs);
    int c = (int)(idx % vecs) * 4;
    int s, d;
    if (e < E1N) { s = cs[e]; d = cd[e] + NT; }
    else         { s = dsrc[e - E1N]; d = ddst[e - E1N]; }
    float4 v = *(const float4*)(x + (size_t)s * C + c);
    float* p = agg + (size_t)d * C + c;
    atomicAdd(p + 0, v.x); atomicAdd(p + 1, v.y);
    atomicAdd(p + 2, v.z); atomicAdd(p + 3, v.w);
}

// edge embeddings: oe[e] = concat(ne[src[e]], ne[dst[e]])
__global__ void k_edges(const float* __restrict__ ne,
                        const int* __restrict__ cs, const int* __restrict__ cd,
                        const int* __restrict__ dsrc, const int* __restrict__ ddst,
                        float* __restrict__ oe) {
    long long idx = (long long)blockIdx.x * blockDim.x + threadIdx.x;
    long long total = (long long)EN * 64;
    if (idx >= total) return;
    int e = (int)(idx >> 6);
    int q = (int)(idx & 63);
    int node, col;
    if (q < 32) {
        node = (e < E1N) ? cs[e] : dsrc[e - E1N];
        col = q * 4;
    } else {
        node = (e < E1N) ? (cd[e] + NT) : ddst[e - E1N];
        col = (q - 32) * 4;
    }
    float4 v = *(const float4*)(ne + (size_t)node * EMB + col);
    *(float4*)(oe + (size_t)e * 256 + q * 4) = v;
}

// graph embedding: column mean of ne [NN, EMB]
__global__ void k_gmean(const float* __restrict__ ne, float* __restrict__ out) {
    __shared__ float red[256];
    int c = blockIdx.x;
    float sum = 0.0f;
    for (int i = threadIdx.x; i < NN; i += 256) sum += ne[(size_t)i * EMB + c];
    red[threadIdx.x] = sum;
    __syncthreads();
    for (int s = 128; s > 0; s >>= 1) {
        if (threadIdx.x < s) red[threadIdx.x] += red[threadIdx.x + s];
        __syncthreads();
    }
    if (threadIdx.x == 0) out[c] = red[0] / (float)NN;
}

// ---------------- WMMA fp32 GEMM ----------------
// Out[M,Nout] = epi( pro(A)[M,K] @ B[K,Nout] + bias )
//   PRO 0: a                PRO 1: relu(a*s[k]+t[k])     PRO 2: a + A2
//   RELU: apply relu in epilogue
// Block: 256 thr = 8 waves; wave tile 16x64 (4x v_wmma_f32_16x16x4_f32 / k-step);
// block tile 128x64. B panel staged in LDS in (k/2, n, k&1) pair layout so each
// lane reads its two B operands as one 8-byte LDS load; pair-row stride 160
// (== 32 mod 64 banks) makes all four b64 reads bank-conflict-free across the
// two half-waves. K chunked by 128. Out-of-range M rows are handled by row
// clamping (no predication in the inner loop); stores are guarded.
template<int PRO, bool RELU>
__global__ __launch_bounds__(256)
void k_gemm(const float* __restrict__ A, const float* __restrict__ A2,
            const float* __restrict__ sv, const float* __restrict__ tv,
            const float* __restrict__ B, const float* __restrict__ bias,
            float* __restrict__ Out, int M, int K, int ldA, int Nout) {
    constexpr int LDB = 160;                  // pair-row stride (floats); 160 % 64 == 32
    __shared__ float Bs[64 * LDB];            // 64 pair-rows = 128 k values (40 KB)
    __shared__ float Ss[256], Ts[256];

    const int tid  = threadIdx.x;
    const int n0   = blockIdx.y * 64;
    const int wave = tid >> 5;
    const int lane = tid & 31;
    const int lh   = lane & 15;               // A row / B,C col within 16
    const int kg   = (lane >> 4) << 1;        // 0 (lanes 0-15) or 2 (lanes 16-31)
    const int hi   = lane >> 4;

    if (PRO == 1) {
        for (int k = tid; k < K; k += 256) { Ss[k] = sv[k]; Ts[k] = tv[k]; }
    }

    const int m = blockIdx.x * 128 + wave * 16 + lh;
    const int m_eff = (m < M) ? m : (M - 1);  // clamp: OOB lanes compute garbage, stores guarded
    const float* Arow  = A  + (size_t)m_eff * ldA;
    const float* A2row = (PRO == 2) ? (A2 + (size_t)m_eff * ldA) : nullptr;

    v8f acc0 = {}, acc1 = {}, acc2 = {}, acc3 = {};

    for (int kc = 0; kc < K; kc += 128) {
        const int kl = (K - kc < 128) ? (K - kc) : 128;
        __syncthreads();
        for (int idx = tid; idx < kl * 64; idx += 256) {
            int k = idx >> 6, n = idx & 63;
            Bs[(k >> 1) * LDB + n * 2 + (k & 1)] = B[(size_t)(kc + k) * Nout + n0 + n];
        }
        __syncthreads();

        #pragma unroll 2
        for (int k0 = 0; k0 < kl; k0 += 4) {
            const int ka = k0 + kg;           // local k for this half-wave
            const int kA = kc + ka;           // global k
            float a0 = Arow[kA];
            float a1 = Arow[kA + 1];
            if (PRO == 1) {
                a0 = fmaxf(fmaf(a0, Ss[kA], Ts[kA]), 0.0f);
                a1 = fmaxf(fmaf(a1, Ss[kA + 1], Ts[kA + 1]), 0.0f);
            } else if (PRO == 2) {
                a0 += A2row[kA];
                a1 += A2row[kA + 1];
            }
            v2f av; av.x = a0; av.y = a1;
            const float* bp = &Bs[(ka >> 1) * LDB + lh * 2];
            v2f b0 = *(const v2f*)(bp);        // n = lh
            v2f b1 = *(const v2f*)(bp + 32);   // n = 16 + lh
            v2f b2 = *(const v2f*)(bp + 64);   // n = 32 + lh
            v2f b3 = *(const v2f*)(bp + 96);   // n = 48 + lh
            acc0 = __builtin_amdgcn_wmma_f32_16x16x4_f32(false, av, false, b0, (short)0, acc0, false, false);
            acc1 = __builtin_amdgcn_wmma_f32_16x16x4_f32(false, av, false, b1, (short)0, acc1, false, false);
            acc2 = __builtin_amdgcn_wmma_f32_16x16x4_f32(false, av, false, b2, (short)0, acc2, false, false);
            acc3 = __builtin_amdgcn_wmma_f32_16x16x4_f32(false, av, false, b3, (short)0, acc3, false, false);
        }
    }

    const float bi0 = bias[n0 + lh];
    const float bi1 = bias[n0 + 16 + lh];
    const float bi2 = bias[n0 + 32 + lh];
    const float bi3 = bias[n0 + 48 + lh];
    #pragma unroll
    for (int r = 0; r < 8; ++r) {
        int row = blockIdx.x * 128 + wave * 16 + hi * 8 + r;
        if (row < M) {
            float* orow = Out + (size_t)row * Nout + n0;
            float o0 = acc0[r] + bi0, o1 = acc1[r] + bi1;
            float o2 = acc2[r] + bi2, o3 = acc3[r] + bi3;
            if (RELU) {
                o0 = fmaxf(o0, 0.0f); o1 = fmaxf(o1, 0.0f);
                o2 = fmaxf(o2, 0.0f); o3 = fmaxf(o3, 0.0f);
            }
            orow[lh] = o0; orow[16 + lh] = o1; orow[32 + lh] = o2; orow[48 + lh] = o3;
        }
    }
}

// ---------------- host orchestration ----------------
extern "C" void kernel_launch(void* const* d_in, const int* in_sizes, int n_in,
                              void* d_out, int out_size, void* d_ws, size_t ws_size,
                              hipStream_t stream) {
    // inputs (setup_inputs order)
    const float* tin[6];  for (int i = 0; i < 6;  ++i) tin[i] = (const float*)d_in[i];
    const float* vin[10]; for (int i = 0; i < 10; ++i) vin[i] = (const float*)d_in[6 + i];
    const int* compat_src = (const int*)d_in[16];
    const int* compat_dst = (const int*)d_in[17];
    const int* dep_src    = (const int*)d_in[18];
    const int* dep_dst    = (const int*)d_in[19];
    const float* te_W1 = (const float*)d_in[20]; const float* te_b1 = (const float*)d_in[21];
    const float* te_g1 = (const float*)d_in[22]; const float* te_be1 = (const float*)d_in[23];
    const float* te_W2 = (const float*)d_in[24]; const float* te_b2 = (const float*)d_in[25];
    const float* te_g2 = (const float*)d_in[26]; const float* te_be2 = (const float*)d_in[27];
    const float* te_W3 = (const float*)d_in[28]; const float* te_b3 = (const float*)d_in[29];
    const float* ve_W1 = (const float*)d_in[30]; const float* ve_b1 = (const float*)d_in[31];
    const float* ve_g1 = (const float*)d_in[32]; const float* ve_be1 = (const float*)d_in[33];
    const float* ve_W2 = (const float*)d_in[34]; const float* ve_b2 = (const float*)d_in[35];
    const float* ve_g2 = (const float*)d_in[36]; const float* ve_be2 = (const float*)d_in[37];
    const float* ve_W3 = (const float*)d_in[38]; const float* ve_b3 = (const float*)d_in[39];
    const float* g1_Wa = (const float*)d_in[40]; const float* g1_ba = (const float*)d_in[41];
    const float* g1_Wb = (const float*)d_in[42]; const float* g1_bb = (const float*)d_in[43];
    const float* g2_Wc = (const float*)d_in[44]; const float* g2_bc = (const float*)d_in[45];
    const float* g2_Wd = (const float*)d_in[46]; const float* g2_bd = (const float*)d_in[47];

    float* out = (float*)d_out;
    float* node_emb = out;                       // [NN, 128]
    float* edge_emb = out + (size_t)NN * EMB;    // [EN, 256]
    float* graph_emb = edge_emb + (size_t)EN * 256; // [128]

    // workspace layout (floats)
    float* ws = (float*)d_ws;
    float* bufA  = ws;                                   // [NN, 256]
    float* bufB  = bufA + (size_t)NN * 256;              // [NN, 256]
    float* nodex = bufB + (size_t)NN * 256;              // [NN, 128]
    float* agg   = nodex + (size_t)NN * EMB;             // [NN, 256]
    float* taskx = agg + (size_t)NN * 256;               // [NT, 8]
    float* vmx   = taskx + (size_t)NT * 8;               // [NV, 12]
    float* w1p   = vmx + (size_t)NV * 12;                // [8, 256]
    float* wv1p  = w1p + 8 * 256;                        // [12, 256]
    float* sbuf  = wv1p + 12 * 256;                      // [256]
    float* tbuf  = sbuf + 256;                           // [256]
    float* maxc  = tbuf + 256;                           // [1]

    const int gNT = (NT + 127) / 128;   // 391
    const int gNV = (NV + 127) / 128;   // 8
    const int gN  = (NN + 127) / 128;   // 399

    // prep
    k_max<<<1, 256, 0, stream>>>(vin[7] /*vm_cpu_cores*/, NV, maxc);
    k_packw<<<(12 * 256 + 255) / 256, 256, 0, stream>>>(te_W1, ve_W1, w1p, wv1p);
    k_feat<<<(NT + 255) / 256, 256, 0, stream>>>(
        tin[0], tin[1], tin[2], tin[3], tin[4], tin[5],
        vin[0], vin[1], vin[2], vin[3], vin[4], vin[5], vin[6], vin[7], vin[8], vin[9],
        taskx, vmx, maxc);

    // ---- task encoder ----
    k_gemm<0, false><<<dim3(gNT, 4), 256, 0, stream>>>(taskx, nullptr, nullptr, nullptr,
                                                       w1p, te_b1, bufA, NT, 8, 8, 256);
    k_colstats<<<256, 256, 0, stream>>>(bufA, NT, 256, te_g1, te_be1, sbuf, tbuf);
    k_gemm<1, false><<<dim3(gNT, 4), 256, 0, stream>>>(bufA, nullptr, sbuf, tbuf,
                                                       te_W2, te_b2, bufB, NT, 256, 256, 256);
    k_colstats<<<256, 256, 0, stream>>>(bufB, NT, 256, te_g2, te_be2, sbuf, tbuf);
    k_gemm<1, false><<<dim3(gNT, 2), 256, 0, stream>>>(bufB, nullptr, sbuf, tbuf,
                                                       te_W3, te_b3, nodex, NT, 256, 256, 128);

    // ---- vm encoder ----
    k_gemm<0, false><<<dim3(gNV, 4), 256, 0, stream>>>(vmx, nullptr, nullptr, nullptr,
                                                       wv1p, ve_b1, bufA, NV, 12, 12, 256);
    k_colstats<<<256, 256, 0, stream>>>(bufA, NV, 256, ve_g1, ve_be1, sbuf, tbuf);
    k_gemm<1, false><<<dim3(gNV, 4), 256, 0, stream>>>(bufA, nullptr, sbuf, tbuf,
                                                       ve_W2, ve_b2, bufB, NV, 256, 256, 256);
    k_colstats<<<256, 256, 0, stream>>>(bufB, NV, 256, ve_g2, ve_be2, sbuf, tbuf);
    k_gemm<1, false><<<dim3(gNV, 2), 256, 0, stream>>>(bufB, nullptr, sbuf, tbuf,
                                                       ve_W3, ve_b3, nodex + (size_t)NT * EMB,
                                                       NV, 256, 256, 128);

    // ---- GIN layer 1 ----
    {
        long long n = (long long)NN * EMB;
        k_fill0<<<(unsigned)((n + 255) / 256), 256, 0, stream>>>(agg, n);
        long long tot = (long long)EN * (EMB / 4);
        k_scatter<<<(unsigned)((tot + 255) / 256), 256, 0, stream>>>(
            nodex, agg, compat_src, compat_dst, dep_src, dep_dst, EMB);
    }
    k_gemm<2, true><<<dim3(gN, 4), 256, 0, stream>>>(nodex, agg, nullptr, nullptr,
                                                     g1_Wa, g1_ba, bufA, NN, 128, 128, 256);
    k_gemm<0, true><<<dim3(gN, 4), 256, 0, stream>>>(bufA, nullptr, nullptr, nullptr,
                                                     g1_Wb, g1_bb, bufB, NN, 256, 256, 256);

    // ---- GIN layer 2 ----
    {
        long long n = (long long)NN * 256;
        k_fill0<<<(unsigned)((n + 255) / 256), 256, 0, stream>>>(agg, n);
        long long tot = (long long)EN * (256 / 4);
        k_scatter<<<(unsigned)((tot + 255) / 256), 256, 0, stream>>>(
            bufB, agg, compat_src, compat_dst, dep_src, dep_dst, 256);
    }
    k_gemm<2, true><<<dim3(gN, 2), 256, 0, stream>>>(bufB, agg, nullptr, nullptr,
                                                     g2_Wc, g2_bc, bufA, NN, 256, 256, 128);
    k_gemm<0, false><<<dim3(gN, 2), 256, 0, stream>>>(bufA, nullptr, nullptr, nullptr,
                                                      g2_Wd, g2_bd, node_emb, NN, 128, 128, 128);

    // ---- outputs ----
    {
        long long tot = (long long)EN * 64;
        k_edges<<<(unsigned)((tot + 255) / 256), 256, 0, stream>>>(
            node_emb, compat_src, compat_dst, dep_src, dep_dst, edge_emb);
    }
    k_gmean<<<128, 256, 0, stream>>>(node_emb, graph_emb);
}